// PFBSynthesisLayer_8924942041336
// MI455X (gfx1250) — compile-verified
//
#include <hip/hip_runtime.h>
#include <hip/hip_bf16.h>

// ---------------------------------------------------------------------------
// PFB synthesis: irfft(256) as f32 WMMA GEMM (M=65536, K=260, N=256),
// then gather-style overlap-add (hop 128, window 768, delay 648).
//
// K-slot ordering (chosen so no chunk mixes Re/Im source arrays):
//   slot 0..127   : Re bin s = slot          (c=1 at s=0, else 2)
//   slot 128      : Re bin s = 128 (Nyquist, c=1)
//   slot 129..131 : zero rows (pad)
//   slot 132..258 : Im bin s = slot-131 (s=1..127, c=2)
//   slot 259      : Im bin s = 128  -> W row forced 0 (irfft ignores Im Nyquist)
// W is stored PAIR-INTERLEAVED: Wp[p*512 + n*2 + (k&1)], p = k>>1, so a lane's
// B fragment {W[k0][n], W[k0+1][n]} is one 8-byte load.
// ---------------------------------------------------------------------------

typedef __attribute__((ext_vector_type(2))) float v2f;
typedef __attribute__((ext_vector_type(8))) float v8f;

#define FFT_LEN   256
#define SPEC_BINS 129           // FFT_LEN/2 + 1
#define KPAD      260           // padded K (65 chunks of 4)
#define NROWS     65536         // B*C*T = 8*2*4096
#define TDIM      4096
#define WIN_LEN   768
#define FRM_LEN   128
#define GROUP_DELAY 648
#define OUTLEN    (TDIM*FRM_LEN - GROUP_DELAY)   // 523640
#define NBC       16            // B*C

// ---------------------------------------------------------------------------
// Kernel 1: build fused, pair-interleaved DFT matrix Wp (KPAD x 256).
// Folds: output reversal (n -> 255-n) and total scale (1/256)*(128*-256) = -128.
// ---------------------------------------------------------------------------
__global__ void build_w_kernel(float* __restrict__ Wp) {
    int k = blockIdx.x;       // 0..KPAD-1 (K slot)
    int n = threadIdx.x;      // 0..255
    float val = 0.0f;
    int s = -1;
    bool isIm = false;
    if (k <= 128)                { s = k;        isIm = false; }  // Re 0..128
    else if (k >= 132 && k <= 258){ s = k - 131; isIm = true;  }  // Im 1..127
    // k in {129,130,131,259}: zero row
    if (s >= 0) {
        float c = (s == 0 || s == 128) ? 1.0f : 2.0f;
        int nrev  = 255 - n;                       // fold output reversal
        int phase = (s * nrev) & 255;              // exact angle reduction
        float ang = (float)phase * 0.0245436926061702596f;   // pi/128
        float sn, cs;
        __sincosf(ang, &sn, &cs);
        val = isIm ? (128.0f * c * sn) : (-128.0f * c * cs);
    }
    // pair-interleaved store: one b64 per lane-fragment in the GEMM
    Wp[(k >> 1) * 512 + n * 2 + (k & 1)] = val;
}

// ---------------------------------------------------------------------------
// Kernel 2: GEMM via V_WMMA_F32_16X16X4_F32.
// One wave computes a 16(t)-row x 256(n)-col tile; 16 f32 accumulators.
//
// A (16x4 f32, ISA 7.12.2): lanes 0-15 -> {K=0,K=1}, lanes 16-31 -> {K=2,K=3};
// M = lane&15.  B (4x16): transpose-symmetric.  C/D: VGPR j -> M=j (lanes
// 0-15) / M=j+8 (lanes 16-31); N = lane&15.
// ---------------------------------------------------------------------------
__global__ void __launch_bounds__(256)
ifft_wmma_kernel(const float* __restrict__ re, const float* __restrict__ im,
                 const float* __restrict__ Wp, float* __restrict__ fft) {
    const int wid  = (blockIdx.x * blockDim.x + threadIdx.x) >> 5;  // 0..4095
    const int lane = threadIdx.x & 31;
    const int half = lane >> 4;     // K pair select
    const int mrow = lane & 15;     // M (A rows) / N (B cols)

    const int r0 = wid << 4;
    const int bc = r0 >> 12;
    const int t  = (r0 & (TDIM - 1)) + mrow;   // 16 consecutive t: contiguous

    // lane-resident base pointers; all loop offsets are compile-time immediates
    const float* __restrict__ pR = re + ((size_t)bc * SPEC_BINS) * TDIM + t
                                      + (size_t)(2 * half) * TDIM;        // Re s = 2*half
    const float* __restrict__ pI = im + ((size_t)bc * SPEC_BINS) * TDIM + t
                                      + (size_t)(1 + 2 * half) * TDIM;    // Im s = 1+2*half
    const float* __restrict__ pW = Wp + (size_t)half * 512 + (size_t)mrow * 2;

    v8f acc[16];
#pragma unroll
    for (int i = 0; i < 16; ++i) acc[i] = (v8f){};

    // ---- Phase R: K slots 0..127 (all Re, branch-free) ----
#pragma unroll 4
    for (int kc = 0; kc < 32; ++kc) {
        v2f a;
        a.x = pR[(size_t)(4 * kc) * TDIM];
        a.y = pR[(size_t)(4 * kc + 1) * TDIM];
        const float* __restrict__ w = pW + (size_t)(2 * kc) * 512;
#pragma unroll
        for (int nt = 0; nt < 16; ++nt) {
            v2f b = *(const v2f*)(w + nt * 32);
            acc[nt] = __builtin_amdgcn_wmma_f32_16x16x4_f32(
                false, a, false, b, (short)0, acc[nt], false, false);
        }
    }

    // ---- Tail chunk: K slots 128..131 (Re Nyquist + zero pad) ----
    {
        v2f a;
        a.x = (half == 0) ? pR[(size_t)(128 - 2 * half) * TDIM] : 0.0f;
        a.y = 0.0f;
        const float* __restrict__ w = pW + (size_t)(2 * 32) * 512;
#pragma unroll
        for (int nt = 0; nt < 16; ++nt) {
            v2f b = *(const v2f*)(w + nt * 32);
            acc[nt] = __builtin_amdgcn_wmma_f32_16x16x4_f32(
                false, a, false, b, (short)0, acc[nt], false, false);
        }
    }

    // ---- Phase I: K slots 132..259 (all Im, branch-free; slot 259 row of W
    //      is zero so the Im-Nyquist load is harmless) ----
#pragma unroll 4
    for (int kc = 0; kc < 32; ++kc) {
        v2f a;
        a.x = pI[(size_t)(4 * kc) * TDIM];
        a.y = pI[(size_t)(4 * kc + 1) * TDIM];
        const float* __restrict__ w = pW + (size_t)(2 * (kc + 33)) * 512;
#pragma unroll
        for (int nt = 0; nt < 16; ++nt) {
            v2f b = *(const v2f*)(w + nt * 32);
            acc[nt] = __builtin_amdgcn_wmma_f32_16x16x4_f32(
                false, a, false, b, (short)0, acc[nt], false, false);
        }
    }

    // ---- Store reversed+scaled irfft rows: fft[r][n], row stride 256 ----
    float* __restrict__ outp = fft + (size_t)r0 * FFT_LEN + mrow;
#pragma unroll
    for (int nt = 0; nt < 16; ++nt) {
#pragma unroll
        for (int j = 0; j < 8; ++j) {
            outp[(size_t)(j + half * 8) * FFT_LEN + nt * 16] = acc[nt][j];
        }
    }
}

// ---------------------------------------------------------------------------
// Kernel 3: overlap-add as a gather (<= 6 taps), scale already in fft.
//   y[bc][n] = sum_t h[k] * fft[bc][t][k & 255],  k = (n+648) - 128*t
// ---------------------------------------------------------------------------
__global__ void overlap_add_kernel(const float* __restrict__ fft,
                                   const float* __restrict__ h,
                                   float* __restrict__ out) {
    const int idx = blockIdx.x * blockDim.x + threadIdx.x;
    if (idx >= NBC * OUTLEN) return;
    const int bc = idx / OUTLEN;
    const int n  = idx - bc * OUTLEN;
    const int m  = n + GROUP_DELAY;

    int tlo = (m - (WIN_LEN - FRM_LEN)) >> 7;   // ceil((m-767)/128), m>=648
    if (tlo < 0) tlo = 0;
    int thi = m >> 7;
    if (thi > TDIM - 1) thi = TDIM - 1;

    const float* __restrict__ f = fft + (size_t)bc * TDIM * FFT_LEN;
    float acc = 0.0f;
    for (int tt = tlo; tt <= thi; ++tt) {
        const int k = m - (tt << 7);            // 0..767
        acc += h[k] * f[tt * FFT_LEN + (k & 255)];
    }
    out[idx] = acc;
}

// ---------------------------------------------------------------------------
extern "C" void kernel_launch(void* const* d_in, const int* in_sizes, int n_in,
                              void* d_out, int out_size, void* d_ws, size_t ws_size,
                              hipStream_t stream) {
    const float* re = (const float*)d_in[0];   // (8,2,129,4096) f32
    const float* im = (const float*)d_in[1];   // (8,2,129,4096) f32
    const float* h  = (const float*)d_in[2];   // (768,) f32
    float* out = (float*)d_out;                // (8,2,523640) f32

    float* Wp  = (float*)d_ws;                 // (KPAD/2)*512 floats
    float* fft = Wp + (size_t)(KPAD / 2) * 512; // NROWS*256 floats (~64MB, L2-resident)

    build_w_kernel<<<KPAD, FFT_LEN, 0, stream>>>(Wp);
    ifft_wmma_kernel<<<(NROWS / 16) * 32 / 256, 256, 0, stream>>>(re, im, Wp, fft);
    const int total = NBC * OUTLEN;
    overlap_add_kernel<<<(total + 255) / 256, 256, 0, stream>>>(fft, h, out);
}